// SAGE_43224550868302
// MI455X (gfx1250) — compile-verified
//
#include <hip/hip_runtime.h>
#include <math.h>

#define NNODES 50000
#define NEDGES 800000
#define DHID   128
#define DOUT   64
#define BNEPS  1e-5f

typedef float v2f __attribute__((ext_vector_type(2)));
typedef float v8f __attribute__((ext_vector_type(8)));

// ---------------------------------------------------------------- utilities
__global__ void sage_zero_i32_kernel(int* __restrict__ p, int n) {
    int i = blockIdx.x * blockDim.x + threadIdx.x;
    if (i < n) p[i] = 0;
}

// Count in-degree (integer atomics; cheap: one per edge)
__global__ void sage_count_kernel(const int* __restrict__ dst,
                                  int* __restrict__ cnt, int E) {
    int e = blockIdx.x * blockDim.x + threadIdx.x;
    if (e < E) atomicAdd(&cnt[e < E ? dst[e] : 0], 1);
}

// Single-block exclusive scan over cnt -> rowptr (n = NNODES), rowptr[n] = E.
__global__ __launch_bounds__(1024) void sage_scan_kernel(
    const int* __restrict__ cnt, int* __restrict__ rowptr, int n) {
    __shared__ int buf[1024];
    __shared__ int carry;
    if (threadIdx.x == 0) carry = 0;
    __syncthreads();
    for (int base = 0; base < n; base += 1024) {
        int i = base + (int)threadIdx.x;
        int v = (i < n) ? cnt[i] : 0;
        buf[threadIdx.x] = v;
        __syncthreads();
        for (int off = 1; off < 1024; off <<= 1) {
            int t = (threadIdx.x >= (unsigned)off) ? buf[threadIdx.x - off] : 0;
            __syncthreads();
            buf[threadIdx.x] += t;
            __syncthreads();
        }
        int excl = buf[threadIdx.x] - v;
        if (i < n) rowptr[i] = carry + excl;
        __syncthreads();
        if (threadIdx.x == 1023) carry += buf[1023];
        __syncthreads();
    }
    if (threadIdx.x == 0) rowptr[n] = carry;
}

// cursor = rowptr copy; invdeg = 1/max(cnt,1)
__global__ void sage_prep_kernel(const int* __restrict__ rowptr,
                                 const int* __restrict__ cnt,
                                 int* __restrict__ cursor,
                                 float* __restrict__ invdeg, int n) {
    int i = blockIdx.x * blockDim.x + threadIdx.x;
    if (i < n) {
        cursor[i] = rowptr[i];
        invdeg[i] = 1.0f / fmaxf((float)cnt[i], 1.0f);
    }
}

// Place src node ids into CSR adjacency (nbr) via per-node cursors.
__global__ void sage_fill_kernel(const int* __restrict__ src,
                                 const int* __restrict__ dst,
                                 int* __restrict__ cursor,
                                 int* __restrict__ nbr, int E) {
    int e = blockIdx.x * blockDim.x + threadIdx.x;
    if (e < E) {
        int pos = atomicAdd(&cursor[dst[e]], 1);
        nbr[pos] = src[e];
    }
}

// Mean aggregation as a gather: one wave per node, lane owns 4 channels.
// agg[n][:] = invdeg[n] * sum_{s in nbr(n)} h[s][:]
__global__ __launch_bounds__(256) void sage_gather_kernel(
    const float* __restrict__ h, const int* __restrict__ rowptr,
    const int* __restrict__ nbr, const float* __restrict__ invdeg,
    float* __restrict__ agg) {
    int node = blockIdx.x * (blockDim.x >> 5) + (threadIdx.x >> 5);
    if (node >= NNODES) return;
    int lane = threadIdx.x & 31;
    int jb = rowptr[node];
    int je = rowptr[node + 1];
    float ax = 0.f, ay = 0.f, az = 0.f, aw = 0.f;
    for (int j = jb; j < je; ++j) {
        int s = nbr[j];
        const float4 v = *(const float4*)(h + (size_t)s * DHID + lane * 4);
        ax += v.x; ay += v.y; az += v.z; aw += v.w;
    }
    float inv = invdeg[node];
    float4 r;
    r.x = ax * inv; r.y = ay * inv; r.z = az * inv; r.w = aw * inv;
    *(float4*)(agg + (size_t)node * DHID + lane * 4) = r;
}

// ------------------------------------------------- fused SAGE layer (WMMA)
// hout = BN( relu?( agg @ WL^T + BL + hin @ WR^T ) )   (agg already mean'ed)
__global__ __launch_bounds__(256) void sage_gemm_bn_kernel(
    const float* __restrict__ agg, const float* __restrict__ hin,
    const float* __restrict__ WL, const float* __restrict__ BL,
    const float* __restrict__ WR,
    const float* __restrict__ G,  const float* __restrict__ BE,
    const float* __restrict__ M,  const float* __restrict__ V,
    float* __restrict__ hout, int din, int dout, int do_relu)
{
    const int wid    = blockIdx.x * (blockDim.x >> 5) + (threadIdx.x >> 5);
    const int ct     = dout >> 4;
    const int ntiles = (NNODES / 16) * ct;
    if (wid >= ntiles) return;                 // wave-uniform exit

    const int r  = wid / ct;
    const int c  = wid % ct;
    const int rb = r << 4;
    const int cb = c << 4;

    const int lane = threadIdx.x & 31;
    const int half = lane >> 4;                // 0: K={0,1}, 1: K={2,3}
    const int l    = lane & 15;
    const int ko   = half * 2;

    v8f acc;
    const float bias = BL[cb + l];
#pragma unroll
    for (int i = 0; i < 8; ++i) acc[i] = bias;

    const int    rA    = rb + l;
    const float* aRow  = agg + (size_t)rA * din + ko;
    const float* hRow  = hin + (size_t)rA * din + ko;
    const float* wlRow = WL  + (size_t)(cb + l) * din + ko;
    const float* wrRow = WR  + (size_t)(cb + l) * din + ko;

#pragma unroll 4
    for (int k = 0; k < din; k += 4) {
        v2f a, b;
        a.x = aRow[k];  a.y = aRow[k + 1];
        b.x = wlRow[k]; b.y = wlRow[k + 1];
        acc = __builtin_amdgcn_wmma_f32_16x16x4_f32(false, a, false, b,
                                                    (short)0, acc, false, false);
    }
#pragma unroll 4
    for (int k = 0; k < din; k += 4) {
        v2f a, b;
        a.x = hRow[k];  a.y = hRow[k + 1];
        b.x = wrRow[k]; b.y = wrRow[k + 1];
        acc = __builtin_amdgcn_wmma_f32_16x16x4_f32(false, a, false, b,
                                                    (short)0, acc, false, false);
    }

    const int   o  = cb + l;
    const float sc = G[o] * rsqrtf(V[o] + BNEPS);
    const float sh = BE[o] - M[o] * sc;
    float* orow = hout + (size_t)(rb + half * 8) * dout + o;
#pragma unroll
    for (int i = 0; i < 8; ++i) {
        float y = acc[i];
        if (do_relu) y = fmaxf(y, 0.0f);
        orow[(size_t)i * dout] = y * sc + sh;
    }
}

// ------------------------------------------------------- final linear head
__global__ void sage_head_kernel(const float* __restrict__ h,
                                 const float* __restrict__ fw,
                                 const float* __restrict__ fb,
                                 float* __restrict__ out, int n) {
    int i = blockIdx.x * blockDim.x + threadIdx.x;
    if (i >= n) return;
    const float* row = h + (size_t)i * DOUT;
    float s = fb[0];
#pragma unroll
    for (int k = 0; k < DOUT; ++k) s = fmaf(row[k], fw[k], s);
    out[i] = 1.0f / (1.0f + expf(-s));
}

// ---------------------------------------------------------------- launcher
extern "C" void kernel_launch(void* const* d_in, const int* in_sizes, int n_in,
                              void* d_out, int out_size, void* d_ws, size_t ws_size,
                              hipStream_t stream) {
    (void)in_sizes; (void)n_in; (void)out_size; (void)ws_size;

    const float* x   = (const float*)d_in[0];
    const int*   ei  = (const int*)d_in[1];
    const int*   src = ei;
    const int*   dst = ei + NEDGES;
    const float* wl[3] = {(const float*)d_in[2],  (const float*)d_in[9],  (const float*)d_in[16]};
    const float* bl[3] = {(const float*)d_in[3],  (const float*)d_in[10], (const float*)d_in[17]};
    const float* wr[3] = {(const float*)d_in[4],  (const float*)d_in[11], (const float*)d_in[18]};
    const float* g[3]  = {(const float*)d_in[5],  (const float*)d_in[12], (const float*)d_in[19]};
    const float* be[3] = {(const float*)d_in[6],  (const float*)d_in[13], (const float*)d_in[20]};
    const float* m[3]  = {(const float*)d_in[7],  (const float*)d_in[14], (const float*)d_in[21]};
    const float* v[3]  = {(const float*)d_in[8],  (const float*)d_in[15], (const float*)d_in[22]};
    const float* fw = (const float*)d_in[23];
    const float* fb = (const float*)d_in[24];
    float* out = (float*)d_out;

    // Workspace layout: ints first, then float buffers (all 16B-aligned slots)
    int*   cnt    = (int*)d_ws;                 // N      (rounded 50176)
    int*   rowptr = cnt + 50176;                // N+1    (rounded 50176)
    int*   cursor = rowptr + 50176;             // N      (rounded 50176)
    int*   nbr    = cursor + 50176;             // E      (800000)
    float* invdeg = (float*)(nbr + NEDGES);     // N      (rounded 50176)
    float* agg    = invdeg + 50176;             // N*128
    float* h1     = agg + (size_t)NNODES * DHID;
    float* h2     = h1  + (size_t)NNODES * DHID;
    float* h3     = h1;                         // layer2 reads h2/agg only

    const int B = 256;
    const int nBlocks = (NNODES + B - 1) / B;
    const int eBlocks = (NEDGES + B - 1) / B;

    // ---- CSR build (once; graph shared by all 3 layers)
    sage_zero_i32_kernel<<<nBlocks, B, 0, stream>>>(cnt, NNODES);
    sage_count_kernel<<<eBlocks, B, 0, stream>>>(dst, cnt, NEDGES);
    sage_scan_kernel<<<1, 1024, 0, stream>>>(cnt, rowptr, NNODES);
    sage_prep_kernel<<<nBlocks, B, 0, stream>>>(rowptr, cnt, cursor, invdeg, NNODES);
    sage_fill_kernel<<<eBlocks, B, 0, stream>>>(src, dst, cursor, nbr, NEDGES);

    const int gatherBlocks = (NNODES + 7) / 8;               // 8 waves / block
    const int tiles128     = (NNODES / 16) * (DHID / 16);    // 25000
    const int tiles64      = (NNODES / 16) * (DOUT / 16);    // 12500
    const int gemmB128     = (tiles128 + 7) / 8;
    const int gemmB64      = (tiles64 + 7) / 8;

    // ---- layer 0 : x -> h1 (ReLU + BN)
    sage_gather_kernel<<<gatherBlocks, B, 0, stream>>>(x, rowptr, nbr, invdeg, agg);
    sage_gemm_bn_kernel<<<gemmB128, B, 0, stream>>>(agg, x,
        wl[0], bl[0], wr[0], g[0], be[0], m[0], v[0], h1, DHID, DHID, 1);

    // ---- layer 1 : h1 -> h2 (ReLU + BN)
    sage_gather_kernel<<<gatherBlocks, B, 0, stream>>>(h1, rowptr, nbr, invdeg, agg);
    sage_gemm_bn_kernel<<<gemmB128, B, 0, stream>>>(agg, h1,
        wl[1], bl[1], wr[1], g[1], be[1], m[1], v[1], h2, DHID, DHID, 1);

    // ---- layer 2 : h2 -> h3 (BN only)
    sage_gather_kernel<<<gatherBlocks, B, 0, stream>>>(h2, rowptr, nbr, invdeg, agg);
    sage_gemm_bn_kernel<<<gemmB64, B, 0, stream>>>(agg, h2,
        wl[2], bl[2], wr[2], g[2], be[2], m[2], v[2], h3, DHID, DOUT, 0);

    // ---- head : sigmoid(h3 @ fw^T + fb)
    sage_head_kernel<<<nBlocks, B, 0, stream>>>(h3, fw, fb, out, NNODES);
}